// NetNew_17162689315115
// MI455X (gfx1250) — compile-verified
//
#include <hip/hip_runtime.h>

typedef float v2f __attribute__((ext_vector_type(2)));
typedef float v8f __attribute__((ext_vector_type(8)));

#define WAVES_PER_BLOCK 4
#define THREADS (WAVES_PER_BLOCK * 32)
#define HSTRIDE 84   // 16 rows x 84 cols per h tile (cols 80..83 = zero K-pad)
#define ZSTRIDE 17   // padded 16x16 z tile
#define W_LDS_TOTAL 5248

__device__ __forceinline__ float clip_mag(float v, float mx) {
    // matches jnp.where(|v|>=mx, v*|mx/v|, v) for finite v
    return (fabsf(v) >= mx) ? copysignf(mx, v) : v;
}

__global__ __launch_bounds__(THREADS)
void symnet_wmma_kernel(const float* __restrict__ x,
                        const float* __restrict__ W1, const float* __restrict__ W2,
                        const float* __restrict__ W3, const float* __restrict__ W4,
                        const float* __restrict__ W5, const float* __restrict__ W6,
                        const float* __restrict__ W7, const float* __restrict__ W8,
                        const float* __restrict__ Wf,
                        float* __restrict__ out, int B)
{
    // per-layer geometry (in_dim grows by 9; h column window slides left)
    constexpr int in_dims[8] = { 8, 17, 26, 35, 44, 53, 62, 71 };
    constexpr int kpad[8]    = { 8, 20, 28, 36, 44, 56, 64, 72 };   // round up to x4
    constexpr int woff[8]    = { 0, 128, 448, 896, 1472, 2176, 3072, 4096 };

    __shared__ float s_w[W_LDS_TOTAL];                  // Wt, zero padded, [K_pad][16]
    __shared__ float s_wf[80];
    __shared__ float s_h[WAVES_PER_BLOCK][16 * HSTRIDE];
    __shared__ float s_z[WAVES_PER_BLOCK][16 * ZSTRIDE];

    const float* Ws[8] = { W1, W2, W3, W4, W5, W6, W7, W8 };

    const int tid = threadIdx.x;

    // ---- stage all weights (transposed + padded) into LDS, shared by block ----
    for (int l = 0; l < 8; ++l) {
        const int idim = in_dims[l];
        const int n_el = 16 * kpad[l];
        const float* __restrict__ W = Ws[l];
        for (int idx = tid; idx < n_el; idx += THREADS) {
            const int k = idx >> 4;        // K index (row of Wt)
            const int n = idx & 15;        // N index (output col, 13 valid)
            float v = 0.0f;
            if (n < 13 && k < idim) v = W[n * idim + k];   // Wt[k][n] = W[n][k]
            s_w[woff[l] + idx] = v;
        }
    }
    if (tid < 80) s_wf[tid] = Wf[tid];
    __syncthreads();

    const int wave = tid >> 5;
    const int lane = tid & 31;
    const int m    = lane & 15;    // row (A) or column (B/C)
    const int half = lane >> 4;    // lane half selects K/M grouping
    float* __restrict__ h = s_h[wave];
    float* __restrict__ z = s_z[wave];

    const long long tileRow = (long long)(blockIdx.x * WAVES_PER_BLOCK + wave) * 16;
    if (tileRow >= B) return;

    // ---- load 16x8 x tile into h columns 72..79 (float4 per lane) ----
    {
        const int r = lane >> 1;
        const int c = (lane & 1) * 4;
        const float4 v = *(const float4*)(x + (size_t)(tileRow + r) * 8 + c);
        float* dst = &h[r * HSTRIDE + 72 + c];
        dst[0] = v.x; dst[1] = v.y; dst[2] = v.z; dst[3] = v.w;
    }
    // zero the K-padding columns 80..83 (A pad values multiply zero B rows)
    if (half == 0) {
        float* dst = &h[m * HSTRIDE + 80];
        dst[0] = 0.0f; dst[1] = 0.0f; dst[2] = 0.0f; dst[3] = 0.0f;
    }

    int s = 72;  // current h window start column
#pragma unroll
    for (int l = 0; l < 8; ++l) {
        const float* __restrict__ wl = &s_w[woff[l]];
        const int Kp = kpad[l];
        v8f c = {};  // accumulator (C starts at zero)

        // Z(16x16) = H(16xK) * Wt(Kx16), K in steps of 4 on the f32 matrix pipe
#pragma unroll
        for (int k = 0; k < Kp; k += 4) {
            v2f a, b;
            // A 16x4 layout: VGPR0 = K (lanes 0-15) / K+2 (lanes 16-31); VGPR1 = K+1 / K+3
            a.x = h[m * HSTRIDE + s + k + 2 * half];
            a.y = h[m * HSTRIDE + s + k + 2 * half + 1];
            // B 4x16 layout: row-striped across lanes; VGPR0 = rows K,K+1; VGPR1 = rows K+2,K+3
            b.x = wl[(k + half) * 16 + m];
            b.y = wl[(k + 2 + half) * 16 + m];
            c = __builtin_amdgcn_wmma_f32_16x16x4_f32(
                    false, a, false, b, (short)0, c, false, false);
        }

        // spill C fragment: lane holds column m, rows r+8*half
#pragma unroll
        for (int r = 0; r < 8; ++r)
            z[(r + 8 * half) * ZSTRIDE + m] = c[r];

        // elementwise op stage: lanes 0..15, one row each (single wave -> DScnt orders LDS)
        if (half == 0) {
            float zv[13];
#pragma unroll
            for (int i = 0; i < 13; ++i) zv[i] = z[m * ZSTRIDE + i];

            float o0 = zv[0] + zv[1];
            float o1 = zv[2] - zv[3];
            float o2 = clip_mag(zv[4] * zv[5], 99999999.0f);
            float den = (zv[7] == 0.0f) ? 0.0001f : zv[7];
            float o3 = clip_mag(zv[6] / den, 9999.0f);
            float o4 = __sinf(zv[8]);
            float o5 = __cosf(zv[9]);
            float o6 = __expf(fminf(zv[10], 17.0f));
            float o7 = __logf(fabsf(zv[11]));
            float o8 = clip_mag(zv[12] * zv[12], 99999999.0f);

            float* dst = &h[m * HSTRIDE + s - 9];   // prepend = grow leftward
            dst[0] = o0; dst[1] = o1; dst[2] = o2; dst[3] = o3; dst[4] = o4;
            dst[5] = o5; dst[6] = o6; dst[7] = o7; dst[8] = o8;
        }
        s -= 9;
    }

    // ---- final projection: out = h(16x80) . Wf(80) ----
    if (half == 0) {
        float acc = 0.0f;
#pragma unroll 8
        for (int j = 0; j < 80; ++j)
            acc = fmaf(h[m * HSTRIDE + j], s_wf[j], acc);
        out[tileRow + m] = acc;
    }
}

extern "C" void kernel_launch(void* const* d_in, const int* in_sizes, int n_in,
                              void* d_out, int out_size, void* d_ws, size_t ws_size,
                              hipStream_t stream) {
    const float* x  = (const float*)d_in[0];
    const float* W1 = (const float*)d_in[1];
    const float* W2 = (const float*)d_in[2];
    const float* W3 = (const float*)d_in[3];
    const float* W4 = (const float*)d_in[4];
    const float* W5 = (const float*)d_in[5];
    const float* W6 = (const float*)d_in[6];
    const float* W7 = (const float*)d_in[7];
    const float* W8 = (const float*)d_in[8];
    const float* Wf = (const float*)d_in[9];
    float* out = (float*)d_out;

    const int B = in_sizes[0] / 8;                 // x is (B, 8)
    const int rowsPerBlock = WAVES_PER_BLOCK * 16; // 64
    const int grid = (B + rowsPerBlock - 1) / rowsPerBlock;

    symnet_wmma_kernel<<<dim3(grid), dim3(THREADS), 0, stream>>>(
        x, W1, W2, W3, W4, W5, W6, W7, W8, Wf, out, B);
}